// DNC_70815420776935
// MI455X (gfx1250) — compile-verified
//
#include <hip/hip_runtime.h>
#include <math.h>

// ---------------- problem constants ----------------
#define EPSV 1e-6f

typedef __attribute__((ext_vector_type(2))) float v2f;
typedef __attribute__((ext_vector_type(8))) float v8f;

// d_out offsets (floats), outputs concatenated in reference return order:
// outs[16,1280], new_h[16,1024], new_c[16,1024], mem[16,2048,64],
// link[16,2048,2048], prec[16,2048], rw[16,4,2048], ww[16,2048], u[16,2048]
constexpr size_t O_OUTS = 0;
constexpr size_t O_H    = 20480;
constexpr size_t O_C    = 36864;
constexpr size_t O_MEM  = 53248;
constexpr size_t O_LINK = 2150400;
constexpr size_t O_PREC = 69259264;
constexpr size_t O_RW   = 69292032;
constexpr size_t O_WW   = 69423104;
constexpr size_t O_U    = 69455872;

// workspace offsets (floats)
constexpr size_t W_CAT  = 0;        // [16,1792]
constexpr size_t W_PRE  = 28672;    // [16,4096]
constexpr size_t W_XI   = 94208;    // [16,471]
constexpr size_t W_RK   = 101744;   // [16,4,64]
constexpr size_t W_RS   = 105840;   // [16,4]
constexpr size_t W_WK   = 105904;   // [16,64]
constexpr size_t W_WSTR = 106928;   // [16]
constexpr size_t W_EV   = 106944;   // [16,64]
constexpr size_t W_WV   = 107968;   // [16,64]
constexpr size_t W_FG   = 108992;   // [16,4]
constexpr size_t W_AG   = 109056;   // [16]
constexpr size_t W_WG   = 109072;   // [16]
constexpr size_t W_RM   = 109088;   // [16,4,3]
constexpr size_t W_WCW  = 109280;   // [16,2048]
constexpr size_t W_ALC  = 142048;   // [16,2048]
constexpr size_t W_FWS  = 174816;   // [16,4,2048]
constexpr size_t W_BWS  = 305888;   // [16,4,2048]

// ---------------- device helpers ----------------
__device__ __forceinline__ float sigf(float x) { return 1.0f / (1.0f + expf(-x)); }
__device__ __forceinline__ float splus(float x) { return (x > 20.0f) ? x : log1pf(expf(x)); }

__device__ __forceinline__ float bred_sum(float v, float* s) {
    int t = threadIdx.x;
    __syncthreads(); s[t] = v; __syncthreads();
    for (int o = blockDim.x >> 1; o > 0; o >>= 1) {
        if (t < o) s[t] += s[t + o];
        __syncthreads();
    }
    float r = s[0]; __syncthreads(); return r;
}
__device__ __forceinline__ float bred_max(float v, float* s) {
    int t = threadIdx.x;
    __syncthreads(); s[t] = v; __syncthreads();
    for (int o = blockDim.x >> 1; o > 0; o >>= 1) {
        if (t < o) s[t] = fmaxf(s[t], s[t + o]);
        __syncthreads();
    }
    float r = s[0]; __syncthreads(); return r;
}

// ---------------- 1) pack cat = [x | last_read | h] : [16,1792] --------------
__global__ void k_pack(const float* __restrict__ x, const float* __restrict__ lr,
                       const float* __restrict__ h, float* __restrict__ cat) {
    int t = blockIdx.x * 256 + threadIdx.x;
    if (t >= 16 * 1792) return;
    int b = t / 1792, k = t - b * 1792;
    float v;
    if (k < 512)       v = x[b * 512 + k];
    else if (k < 768)  v = lr[b * 256 + (k - 512)];
    else               v = h[b * 1024 + (k - 768)];
    cat[t] = v;
}

// ---------------- 2) pre = cat @ W_full + b_full  (WMMA f32 16x16x4) ---------
// cat [16,1792], W_full [1792,4096] row-major. One wave per 16-col N tile.
__global__ void k_gemm1(const float* __restrict__ cat, const float* __restrict__ Wf,
                        const float* __restrict__ bfull, float* __restrict__ pre) {
    int wave = threadIdx.x >> 5, lane = threadIdx.x & 31;
    int tile = blockIdx.x * 8 + wave;       // 256 tiles
    int n0 = tile * 16;
    int hi = lane >> 4, rlo = lane & 15;

    v8f acc;
#pragma unroll
    for (int t = 0; t < 8; ++t) acc[t] = 0.0f;

    for (int k0 = 0; k0 < 1792; k0 += 4) {
        const float* ap = cat + rlo * 1792 + k0 + 2 * hi;
        v2f a; a.x = ap[0]; a.y = ap[1];
        int kb = k0 + 2 * hi;
        const float* bp = Wf + (size_t)kb * 4096 + n0 + rlo;
        v2f bv; bv.x = bp[0]; bv.y = bp[4096];
        if ((k0 & 63) == 0 && k0 + 64 < 1792)
            __builtin_prefetch(bp + (size_t)64 * 4096, 0, 1);
        acc = __builtin_amdgcn_wmma_f32_16x16x4_f32(
            false, a, false, bv, (short)0, acc, false, false);
    }
    int col = n0 + rlo;
    int rb = hi * 8;
    float bias = bfull[col];
#pragma unroll
    for (int r = 0; r < 8; ++r)
        pre[(size_t)(rb + r) * 4096 + col] = acc[r] + bias;
}

// ---------------- 3) LN-LSTM controller ----------------
__global__ void k_lnlstm(const float* __restrict__ pre, const float* __restrict__ hin,
                         const float* __restrict__ cin,
                         const float* __restrict__ g1, const float* __restrict__ b1,
                         const float* __restrict__ g2, const float* __restrict__ b2,
                         float* __restrict__ out_h, float* __restrict__ out_c,
                         float* __restrict__ outs) {
    __shared__ float sred[256];
    int b = blockIdx.x, tid = threadIdx.x;
    const float* prow = pre + (size_t)b * 4096;
    float acts[4][4];
#pragma unroll
    for (int g = 0; g < 4; ++g) {
        float xv[4]; float s = 0.f, s2 = 0.f;
#pragma unroll
        for (int q = 0; q < 4; ++q) {
            xv[q] = prow[g * 1024 + tid + 256 * q];
            s += xv[q]; s2 += xv[q] * xv[q];
        }
        float tot = bred_sum(s, sred);
        float tot2 = bred_sum(s2, sred);
        float mu = tot * (1.0f / 1024.0f);
        float var = tot2 * (1.0f / 1024.0f) - mu * mu;
        float rstd = rsqrtf(var + 1e-5f);
#pragma unroll
        for (int q = 0; q < 4; ++q) {
            int idx = g * 1024 + tid + 256 * q;
            acts[g][q] = (xv[q] - mu) * rstd * g1[idx] + b1[idx];
        }
    }
    float nc[4], ov[4];
    float s = 0.f, s2 = 0.f;
#pragma unroll
    for (int q = 0; q < 4; ++q) {
        int hidx = tid + 256 * q;
        float iv = sigf(acts[0][q]);
        float jv = tanhf(acts[1][q]);
        float fv = sigf(acts[2][q] + 1.0f);   // F_BIAS
        ov[q] = sigf(acts[3][q]);
        float c0 = cin[b * 1024 + hidx];
        nc[q] = c0 * fv + iv * jv;
        s += nc[q]; s2 += nc[q] * nc[q];
    }
    float tot = bred_sum(s, sred);
    float tot2 = bred_sum(s2, sred);
    float mu = tot * (1.0f / 1024.0f);
    float var = tot2 * (1.0f / 1024.0f) - mu * mu;
    float rstd = rsqrtf(var + 1e-5f);
#pragma unroll
    for (int q = 0; q < 4; ++q) {
        int hidx = tid + 256 * q;
        float ncn = (nc[q] - mu) * rstd * g2[hidx] + b2[hidx];
        float nh = tanhf(ncn) * ov[q];
        float h0 = hin[b * 1024 + hidx];
        float c0 = cin[b * 1024 + hidx];
        nh = 0.7f * nh + 0.3f * h0;                     // zoneout h
        float ncz = 0.7f * ncn + 0.3f * c0;             // zoneout c
        out_h[b * 1024 + hidx] = nh;
        out_c[b * 1024 + hidx] = ncz;
        outs[b * 1280 + hidx] = nh;                     // outs[:, :H] = out
    }
}

// ---------------- 4) xi = out @ W_if + b_if  (WMMA, N=471 with guards) -------
__global__ void k_gemm2(const float* __restrict__ outh, const float* __restrict__ Wif,
                        const float* __restrict__ bif, float* __restrict__ xi) {
    int wave = threadIdx.x >> 5, lane = threadIdx.x & 31;
    int tile = blockIdx.x * 8 + wave;       // 30 valid tiles
    if (tile >= 30) return;                 // wave-uniform
    int n0 = tile * 16;
    int hi = lane >> 4, rlo = lane & 15;
    int bcol = n0 + rlo;
    bool bok = bcol < 471;

    v8f acc;
#pragma unroll
    for (int t = 0; t < 8; ++t) acc[t] = 0.0f;

    for (int k0 = 0; k0 < 1024; k0 += 4) {
        const float* ap = outh + rlo * 1024 + k0 + 2 * hi;
        v2f a; a.x = ap[0]; a.y = ap[1];
        int kb = k0 + 2 * hi;
        v2f bv;
        bv.x = bok ? Wif[(size_t)kb * 471 + bcol] : 0.0f;
        bv.y = bok ? Wif[(size_t)(kb + 1) * 471 + bcol] : 0.0f;
        acc = __builtin_amdgcn_wmma_f32_16x16x4_f32(
            false, a, false, bv, (short)0, acc, false, false);
    }
    int col = n0 + rlo;
    int rb = hi * 8;
    if (col < 471) {
        float bias = bif[col];
#pragma unroll
        for (int r = 0; r < 8; ++r)
            xi[(size_t)(rb + r) * 471 + col] = acc[r] + bias;
    }
}

// ---------------- 5) interface vector split + activations --------------------
__global__ void k_iface(const float* __restrict__ xi,
                        float* __restrict__ rk, float* __restrict__ rs,
                        float* __restrict__ wk, float* __restrict__ wstr,
                        float* __restrict__ ev, float* __restrict__ wv,
                        float* __restrict__ fg, float* __restrict__ ag,
                        float* __restrict__ wg, float* __restrict__ rm) {
    int b = blockIdx.x, tid = threadIdx.x;
    const float* xr = xi + b * 471;
    for (int p = tid; p < 459; p += 128) {
        float v = xr[p];
        if (p < 256)        rk[b * 256 + p] = tanhf(v);
        else if (p < 260)   rs[b * 4 + (p - 256)] = splus(v);
        else if (p < 324)   wk[b * 64 + (p - 260)] = tanhf(v);
        else if (p == 324)  wstr[b] = splus(v);
        else if (p < 389)   ev[b * 64 + (p - 325)] = sigf(v);
        else if (p < 453)   wv[b * 64 + (p - 389)] = tanhf(v);
        else if (p < 457)   fg[b * 4 + (p - 453)] = sigf(v);
        else if (p == 457)  ag[b] = sigf(v);
        else                wg[b] = sigf(v);
    }
    if (tid < 4) { // read-mode softmax over 3
        float v0 = xr[459 + tid * 3 + 0];
        float v1 = xr[459 + tid * 3 + 1];
        float v2 = xr[459 + tid * 3 + 2];
        float mx = fmaxf(v0, fmaxf(v1, v2));
        float e0 = expf(v0 - mx), e1 = expf(v1 - mx), e2 = expf(v2 - mx);
        float inv = 1.0f / (e0 + e1 + e2);
        rm[b * 12 + tid * 3 + 0] = e0 * inv;
        rm[b * 12 + tid * 3 + 1] = e1 * inv;
        rm[b * 12 + tid * 3 + 2] = e2 * inv;
    }
}

// ---------------- 6) usage retention + write content softmax -----------------
__global__ void k_usage_wcw(const float* __restrict__ usage, const float* __restrict__ wwold,
                            const float* __restrict__ rwold, const float* __restrict__ fg,
                            const float* __restrict__ wk, const float* __restrict__ wstr,
                            const float* __restrict__ memory,
                            float* __restrict__ u_out, float* __restrict__ wcw) {
    __shared__ float sred[256];
    __shared__ __align__(16) float wks[64];
    int b = blockIdx.x, tid = threadIdx.x;
    if (tid < 64) wks[tid] = wk[b * 64 + tid];
    __syncthreads();
    float fg0 = fg[b * 4 + 0], fg1 = fg[b * 4 + 1], fg2 = fg[b * 4 + 2], fg3 = fg[b * 4 + 3];
    float nk = 0.f;
#pragma unroll
    for (int k = 0; k < 64; ++k) nk += wks[k] * wks[k];
    nk = sqrtf(nk);
    float beta = wstr[b];
    const float4* wk4 = (const float4*)wks;
    float lg[8];
#pragma unroll
    for (int q = 0; q < 8; ++q) {
        int m = tid + 256 * q;
        float us = usage[b * 2048 + m];
        float u = us + (1.0f - us) * wwold[b * 2048 + m];
        float psi = (1.0f - fg0 * rwold[b * 8192 + m]) *
                    (1.0f - fg1 * rwold[b * 8192 + 2048 + m]) *
                    (1.0f - fg2 * rwold[b * 8192 + 4096 + m]) *
                    (1.0f - fg3 * rwold[b * 8192 + 6144 + m]);
        u *= psi;
        u_out[b * 2048 + m] = u;
        const float4* mr4 = (const float4*)(memory + ((size_t)b * 2048 + m) * 64);
        float nm = 0.f, dot = 0.f;
#pragma unroll
        for (int k4 = 0; k4 < 16; ++k4) {
            float4 mv = mr4[k4];
            float4 kv = wk4[k4];
            nm += mv.x * mv.x + mv.y * mv.y + mv.z * mv.z + mv.w * mv.w;
            dot += mv.x * kv.x + mv.y * kv.y + mv.z * kv.z + mv.w * kv.w;
        }
        nm = sqrtf(nm);
        lg[q] = dot / ((nm + EPSV) * (nk + EPSV)) * beta;
    }
    float mx = -3.4e38f;
#pragma unroll
    for (int q = 0; q < 8; ++q) mx = fmaxf(mx, lg[q]);
    mx = bred_max(mx, sred);
    float s = 0.f;
#pragma unroll
    for (int q = 0; q < 8; ++q) s += expf(lg[q] - mx);
    s = bred_sum(s, sred);
    float inv = 1.0f / s;
#pragma unroll
    for (int q = 0; q < 8; ++q)
        wcw[b * 2048 + tid + 256 * q] = expf(lg[q] - mx) * inv;
}

// ---------------- 7) allocation: bitonic sort + product scan in LDS ----------
__global__ void k_alloc(const float* __restrict__ u_in, float* __restrict__ alloc) {
    __shared__ float vals[2048];
    __shared__ int   idxs[2048];
    __shared__ float sc[2048];
    int b = blockIdx.x, tid = threadIdx.x;   // 1024 threads
#pragma unroll
    for (int t = 0; t < 2; ++t) {
        int e = tid + t * 1024;
        float u = u_in[b * 2048 + e];
        vals[e] = 1e-6f + (1.0f - 1e-6f) * u;
        idxs[e] = e;
    }
    __syncthreads();
    for (int k = 2; k <= 2048; k <<= 1) {
        for (int j = k >> 1; j > 0; j >>= 1) {
#pragma unroll
            for (int t = 0; t < 2; ++t) {
                int e = tid + t * 1024;
                int x = e ^ j;
                if (x > e) {
                    bool up = ((e & k) == 0);
                    float ve = vals[e], vx = vals[x];
                    if ((ve > vx) == up) {
                        vals[e] = vx; vals[x] = ve;
                        int ti = idxs[e]; idxs[e] = idxs[x]; idxs[x] = ti;
                    }
                }
            }
            __syncthreads();
        }
    }
    // inclusive product scan (Hillis-Steele), then exclusive
    sc[tid] = vals[tid]; sc[tid + 1024] = vals[tid + 1024];
    __syncthreads();
    for (int off = 1; off < 2048; off <<= 1) {
        float r0 = (tid >= off) ? sc[tid - off] * sc[tid] : sc[tid];
        int e2 = tid + 1024;
        float r1 = sc[e2 - off] * sc[e2];   // e2 >= 1024 >= off always
        __syncthreads();
        sc[tid] = r0; sc[e2] = r1;
        __syncthreads();
    }
#pragma unroll
    for (int t = 0; t < 2; ++t) {
        int pos = tid + t * 1024;
        float su = vals[pos];
        float ex = (pos == 0) ? 1.0f : sc[pos - 1];
        alloc[b * 2048 + idxs[pos]] = (1.0f - su) * ex;   // unsort via scatter
    }
}

// ---------------- 8) write weights + precedence ------------------------------
__global__ void k_write(const float* __restrict__ alloc, const float* __restrict__ wcw,
                        const float* __restrict__ ag, const float* __restrict__ wg,
                        const float* __restrict__ prec_old,
                        float* __restrict__ ww_out, float* __restrict__ prec_out) {
    __shared__ float sred[256];
    int b = blockIdx.x, tid = threadIdx.x;
    float a = ag[b], g = wg[b];
    float wwv[8]; float s = 0.f;
#pragma unroll
    for (int q = 0; q < 8; ++q) {
        int m = tid + 256 * q;
        wwv[q] = g * (a * alloc[b * 2048 + m] + (1.0f - a) * wcw[b * 2048 + m]);
        ww_out[b * 2048 + m] = wwv[q];
        s += wwv[q];
    }
    float tot = bred_sum(s, sred);
#pragma unroll
    for (int q = 0; q < 8; ++q) {
        int m = tid + 256 * q;
        prec_out[b * 2048 + m] = (1.0f - tot) * prec_old[b * 2048 + m] + wwv[q];
    }
}

// ---------------- 9) memory update (b128 streaming) --------------------------
__global__ void k_memupd(const float* __restrict__ memory, const float* __restrict__ ww,
                         const float* __restrict__ ev, const float* __restrict__ wv,
                         float* __restrict__ mem_out) {
    int idx4 = blockIdx.x * 256 + threadIdx.x;   // 524288 float4's
    int b = idx4 >> 15;
    int m = (idx4 >> 4) & 2047;
    int wq = idx4 & 15;
    float wm = ww[b * 2048 + m];
    float4 mv = ((const float4*)memory)[idx4];
    float4 evv = ((const float4*)(ev + b * 64))[wq];
    float4 wvv = ((const float4*)(wv + b * 64))[wq];
    float4 o;
    o.x = mv.x * (1.0f - wm * evv.x) + wm * wvv.x;
    o.y = mv.y * (1.0f - wm * evv.y) + wm * wvv.y;
    o.z = mv.z * (1.0f - wm * evv.z) + wm * wvv.z;
    o.w = mv.w * (1.0f - wm * evv.w) + wm * wvv.w;
    ((float4*)mem_out)[idx4] = o;
}

// ------- 10) fused link update + bw (b128 streaming, register col-sums) ------
// block = (b, j-tile of 1024). Thread owns 4 consecutive columns, loops rows.
__global__ void k_linkbw(const float* __restrict__ link_old, const float* __restrict__ ww,
                         const float* __restrict__ prec_old, const float* __restrict__ rwold,
                         float* __restrict__ link_new, float* __restrict__ bw) {
    int b = blockIdx.x, jt = blockIdx.y;         // jt in {0,1}
    int j0 = jt * 1024 + threadIdx.x * 4;
    const float4* Lb = (const float4*)(link_old + (size_t)b * 2048 * 2048 + j0);
    float4*       Ln = (float4*)(link_new + (size_t)b * 2048 * 2048 + j0);
    float4 wwj = *(const float4*)(ww + b * 2048 + j0);
    float4 pj  = *(const float4*)(prec_old + b * 2048 + j0);
    const float* rwb = rwold + b * 8192;
    float acc[4][4];
#pragma unroll
    for (int r = 0; r < 4; ++r)
#pragma unroll
        for (int c = 0; c < 4; ++c) acc[r][c] = 0.0f;

    for (int i = 0; i < 2048; ++i) {
        float wwi = ww[b * 2048 + i];            // wave-uniform -> SGPR
        float4 l = Lb[(size_t)i * 512];
        float4 v;
        v.x = (1.0f - wwi - wwj.x) * l.x + wwi * pj.x;
        v.y = (1.0f - wwi - wwj.y) * l.y + wwi * pj.y;
        v.z = (1.0f - wwi - wwj.z) * l.z + wwi * pj.z;
        v.w = (1.0f - wwi - wwj.w) * l.w + wwi * pj.w;
        if (i == j0 + 0) v.x = 0.0f;             // zero diagonal
        if (i == j0 + 1) v.y = 0.0f;
        if (i == j0 + 2) v.z = 0.0f;
        if (i == j0 + 3) v.w = 0.0f;
        Ln[(size_t)i * 512] = v;
        float r0 = rwb[i], r1 = rwb[2048 + i], r2 = rwb[4096 + i], r3 = rwb[6144 + i];
        acc[0][0] += r0 * v.x; acc[0][1] += r0 * v.y; acc[0][2] += r0 * v.z; acc[0][3] += r0 * v.w;
        acc[1][0] += r1 * v.x; acc[1][1] += r1 * v.y; acc[1][2] += r1 * v.z; acc[1][3] += r1 * v.w;
        acc[2][0] += r2 * v.x; acc[2][1] += r2 * v.y; acc[2][2] += r2 * v.z; acc[2][3] += r2 * v.w;
        acc[3][0] += r3 * v.x; acc[3][1] += r3 * v.y; acc[3][2] += r3 * v.z; acc[3][3] += r3 * v.w;
    }
#pragma unroll
    for (int r = 0; r < 4; ++r) {
        float4 o; o.x = acc[r][0]; o.y = acc[r][1]; o.z = acc[r][2]; o.w = acc[r][3];
        *(float4*)(bw + b * 8192 + r * 2048 + j0) = o;
    }
}

// ---------------- 11) fw = link_new @ rw_old^T (b128 row sums) ---------------
__global__ void k_fw(const float* __restrict__ link_new, const float* __restrict__ rwold,
                     float* __restrict__ fw) {
    __shared__ float sred[256];
    int b = blockIdx.x >> 7;
    int i0 = (blockIdx.x & 127) * 16;
    int tid = threadIdx.x;
    const float* Ln = link_new + (size_t)b * 2048 * 2048;
    const float4* rw4 = (const float4*)(rwold + b * 8192);
    for (int ii = 0; ii < 16; ++ii) {
        int i = i0 + ii;
        const float4* row4 = (const float4*)(Ln + (size_t)i * 2048);
        float f0 = 0.f, f1 = 0.f, f2 = 0.f, f3 = 0.f;
#pragma unroll
        for (int q = 0; q < 2; ++q) {
            int j4 = tid + 256 * q;
            float4 v = row4[j4];
            float4 w0 = rw4[j4];
            float4 w1 = rw4[512 + j4];
            float4 w2 = rw4[1024 + j4];
            float4 w3 = rw4[1536 + j4];
            f0 += v.x * w0.x + v.y * w0.y + v.z * w0.z + v.w * w0.w;
            f1 += v.x * w1.x + v.y * w1.y + v.z * w1.z + v.w * w1.w;
            f2 += v.x * w2.x + v.y * w2.y + v.z * w2.z + v.w * w2.w;
            f3 += v.x * w3.x + v.y * w3.y + v.z * w3.z + v.w * w3.w;
        }
        float t0 = bred_sum(f0, sred);
        float t1 = bred_sum(f1, sred);
        float t2 = bred_sum(f2, sred);
        float t3 = bred_sum(f3, sred);
        if (tid == 0) {
            fw[b * 8192 + i] = t0;
            fw[b * 8192 + 2048 + i] = t1;
            fw[b * 8192 + 4096 + i] = t2;
            fw[b * 8192 + 6144 + i] = t3;
        }
    }
}

// ------- 12) read content softmax + read-weight mix + read vectors -----------
__global__ void k_read(const float* __restrict__ rk, const float* __restrict__ rs,
                       const float* __restrict__ rm, const float* __restrict__ mem_new,
                       const float* __restrict__ fw, const float* __restrict__ bw,
                       float* __restrict__ rw_out, float* __restrict__ outs) {
    __shared__ __align__(16) float rkn[4][64];
    __shared__ float rsL[4];
    __shared__ float rmL[12];
    __shared__ float lg[4][2048];
    __shared__ float sred[256];
    int b = blockIdx.x, tid = threadIdx.x;
    {   // normalized read keys
        int r = tid >> 6, w = tid & 63;
        const float* kr = rk + b * 256 + r * 64;
        float s = 0.f;
        for (int k = 0; k < 64; ++k) s += kr[k] * kr[k];
        rkn[r][w] = kr[w] / (sqrtf(s) + EPSV);
    }
    if (tid < 4) rsL[tid] = rs[b * 4 + tid];
    if (tid < 12) rmL[tid] = rm[b * 12 + tid];
    __syncthreads();

    const float* Mb = mem_new + (size_t)b * 2048 * 64;
    const float4* k0 = (const float4*)&rkn[0][0];
    const float4* k1 = (const float4*)&rkn[1][0];
    const float4* k2 = (const float4*)&rkn[2][0];
    const float4* k3 = (const float4*)&rkn[3][0];
#pragma unroll 1
    for (int q = 0; q < 8; ++q) {
        int m = tid + 256 * q;
        const float4* mr4 = (const float4*)(Mb + (size_t)m * 64);
        float nm = 0.f, d0 = 0.f, d1 = 0.f, d2 = 0.f, d3 = 0.f;
#pragma unroll
        for (int kq = 0; kq < 16; ++kq) {
            float4 v = mr4[kq];
            nm += v.x * v.x + v.y * v.y + v.z * v.z + v.w * v.w;
            float4 a = k0[kq]; d0 += v.x * a.x + v.y * a.y + v.z * a.z + v.w * a.w;
            float4 bq = k1[kq]; d1 += v.x * bq.x + v.y * bq.y + v.z * bq.z + v.w * bq.w;
            float4 cq = k2[kq]; d2 += v.x * cq.x + v.y * cq.y + v.z * cq.z + v.w * cq.w;
            float4 dq = k3[kq]; d3 += v.x * dq.x + v.y * dq.y + v.z * dq.z + v.w * dq.w;
        }
        float inv = 1.0f / (sqrtf(nm) + EPSV);
        lg[0][m] = d0 * inv * rsL[0];
        lg[1][m] = d1 * inv * rsL[1];
        lg[2][m] = d2 * inv * rsL[2];
        lg[3][m] = d3 * inv * rsL[3];
    }
    __syncthreads();
    for (int r = 0; r < 4; ++r) {   // softmax over M per read head
        float mx = -3.4e38f;
#pragma unroll
        for (int q = 0; q < 8; ++q) mx = fmaxf(mx, lg[r][tid + 256 * q]);
        mx = bred_max(mx, sred);
        float s = 0.f;
#pragma unroll
        for (int q = 0; q < 8; ++q) s += expf(lg[r][tid + 256 * q] - mx);
        s = bred_sum(s, sred);
        float inv = 1.0f / s;
#pragma unroll
        for (int q = 0; q < 8; ++q) {
            int m = tid + 256 * q;
            lg[r][m] = expf(lg[r][m] - mx) * inv;
        }
    }
    __syncthreads();
    for (int r = 0; r < 4; ++r) {   // rw_new = rm0*bw + rm1*fw + rm2*cw
#pragma unroll
        for (int q = 0; q < 8; ++q) {
            int m = tid + 256 * q;
            float v = rmL[r * 3 + 0] * bw[b * 8192 + r * 2048 + m]
                    + rmL[r * 3 + 1] * fw[b * 8192 + r * 2048 + m]
                    + rmL[r * 3 + 2] * lg[r][m];
            rw_out[b * 8192 + r * 2048 + m] = v;
            lg[r][m] = v;
        }
    }
    __syncthreads();
    {   // read vectors: thread <-> (r, w)
        int r = tid >> 6, w = tid & 63;
        float acc = 0.f;
        for (int m = 0; m < 2048; ++m)
            acc += lg[r][m] * Mb[(size_t)m * 64 + w];
        outs[b * 1280 + 1024 + r * 64 + w] = acc;
    }
}

// ---------------- host launch ----------------
extern "C" void kernel_launch(void* const* d_in, const int* in_sizes, int n_in,
                              void* d_out, int out_size, void* d_ws, size_t ws_size,
                              hipStream_t stream) {
    const float* x         = (const float*)d_in[0];
    const float* h         = (const float*)d_in[1];
    const float* c         = (const float*)d_in[2];
    const float* last_read = (const float*)d_in[3];
    const float* memory    = (const float*)d_in[4];
    const float* link      = (const float*)d_in[5];
    const float* prec      = (const float*)d_in[6];
    const float* rw_old    = (const float*)d_in[7];
    const float* ww_old    = (const float*)d_in[8];
    const float* usage     = (const float*)d_in[9];
    const float* W_full    = (const float*)d_in[10];
    const float* b_full    = (const float*)d_in[11];
    const float* g1        = (const float*)d_in[12];
    const float* b1        = (const float*)d_in[13];
    const float* g2        = (const float*)d_in[14];
    const float* b2        = (const float*)d_in[15];
    const float* W_if      = (const float*)d_in[16];
    const float* b_if      = (const float*)d_in[17];

    float* out = (float*)d_out;
    float* ws  = (float*)d_ws;

    float* cat  = ws + W_CAT;
    float* pre  = ws + W_PRE;
    float* xi   = ws + W_XI;
    float* rk   = ws + W_RK;
    float* rs   = ws + W_RS;
    float* wk   = ws + W_WK;
    float* wstr = ws + W_WSTR;
    float* ev   = ws + W_EV;
    float* wv   = ws + W_WV;
    float* fg   = ws + W_FG;
    float* ag   = ws + W_AG;
    float* wg   = ws + W_WG;
    float* rm   = ws + W_RM;
    float* wcw  = ws + W_WCW;
    float* alc  = ws + W_ALC;
    float* fwb  = ws + W_FWS;
    float* bwb  = ws + W_BWS;

    k_pack<<<112, 256, 0, stream>>>(x, last_read, h, cat);
    k_gemm1<<<32, 256, 0, stream>>>(cat, W_full, b_full, pre);
    k_lnlstm<<<16, 256, 0, stream>>>(pre, h, c, g1, b1, g2, b2,
                                     out + O_H, out + O_C, out + O_OUTS);
    k_gemm2<<<4, 256, 0, stream>>>(out + O_H, W_if, b_if, xi);
    k_iface<<<16, 128, 0, stream>>>(xi, rk, rs, wk, wstr, ev, wv, fg, ag, wg, rm);
    k_usage_wcw<<<16, 256, 0, stream>>>(usage, ww_old, rw_old, fg, wk, wstr, memory,
                                        out + O_U, wcw);
    k_alloc<<<16, 1024, 0, stream>>>(out + O_U, alc);
    k_write<<<16, 256, 0, stream>>>(alc, wcw, ag, wg, prec, out + O_WW, out + O_PREC);
    k_memupd<<<2048, 256, 0, stream>>>(memory, out + O_WW, ev, wv, out + O_MEM);
    k_linkbw<<<dim3(16, 2), 256, 0, stream>>>(link, out + O_WW, prec, rw_old,
                                              out + O_LINK, bwb);
    k_fw<<<2048, 256, 0, stream>>>(out + O_LINK, rw_old, fwb);
    k_read<<<16, 256, 0, stream>>>(rk, rs, rm, out + O_MEM, fwb, bwb,
                                   out + O_RW, out + O_OUTS);
}